// VectorizedIntelligentNetwork_58196806860976
// MI455X (gfx1250) — compile-verified
//
#include <hip/hip_runtime.h>
#include <math.h>

// Problem dims (compile-time constants from the reference)
#define B_      8
#define T_      16
#define INPUT_  64
#define OUTPUT_ 128
#define N_IN_   32
#define N_TR_   192
#define N_ACT_  32
#define N_      256
#define H_      256
#define KQ_     256
#define S_      256

typedef float v2f __attribute__((ext_vector_type(2)));
typedef float v8f __attribute__((ext_vector_type(8)));

typedef __attribute__((address_space(1))) int global_i32;
typedef __attribute__((address_space(3))) int lds_i32;

// ---- CDNA5 async global->LDS staging (guarded; falls back to plain loads) ---
#if defined(__has_builtin)
#if __has_builtin(__builtin_amdgcn_global_load_async_to_lds_b32)
#define HAVE_ASYNC_LDS 1
#endif
#endif

__device__ __forceinline__ void async_wait_zero() {
#if defined(__has_builtin)
#if __has_builtin(__builtin_amdgcn_s_wait_asynccnt)
    __builtin_amdgcn_s_wait_asynccnt(0);
#else
    asm volatile("s_wait_asynccnt 0x0" ::: "memory");
#endif
#else
    asm volatile("s_wait_asynccnt 0x0" ::: "memory");
#endif
}

// ---------------------------------------------------------------------------
// Per-neuron batched GEMM using V_WMMA_F32_16X16X4_F32.
//   out[b, n, j] (+)= sum_k A[b, n, k] * W[n, k, j] + bias[n, j]
// Batch (M) padded 8 -> 16. One wave computes one 16x16 output tile.
// Block = 256 threads = 8 waves; blockIdx.x = neuron, blockIdx.y = tile group.
// Everything (dims + strides) is a template parameter so the weight stream
// compiles to immediate-offset global_load_b32 off one base pointer.
// ---------------------------------------------------------------------------
template <int Kdim, int Nout, int ASB, int ASN, int OSB, int OSN,
          bool ACC, bool RELU>
__global__ __launch_bounds__(256) void neuron_gemm_wmma(
    const float* __restrict__ A, const float* __restrict__ W,
    const float* __restrict__ bias, float* __restrict__ out)
{
    __shared__ float As[16 * Kdim];   // 16 rows x Kdim (rows 8..15 zero-padded)

    const int n   = blockIdx.x;
    const int tid = threadIdx.x;

    // ---- Stage activation tile into LDS (uniform trip counts; EXEC all-1s) --
#if defined(HAVE_ASYNC_LDS)
    for (int idx = tid; idx < B_ * Kdim; idx += 256) {
        const int m = idx / Kdim;
        const int k = idx - m * Kdim;
        float* src = const_cast<float*>(A) + m * ASB + n * ASN + k;
        __builtin_amdgcn_global_load_async_to_lds_b32(
            (global_i32*)src, (lds_i32*)&As[idx], 0, 0);
    }
    for (int idx = tid; idx < (16 - B_) * Kdim; idx += 256)
        As[B_ * Kdim + idx] = 0.0f;
    async_wait_zero();
    __syncthreads();
#else
    for (int idx = tid; idx < 16 * Kdim; idx += 256) {
        const int m = idx / Kdim;
        const int k = idx - m * Kdim;
        As[idx] = (m < B_) ? A[m * ASB + n * ASN + k] : 0.0f;
    }
    __syncthreads();
#endif

    const int lane  = tid & 31;
    const int wave  = tid >> 5;
    const int col   = lane & 15;          // M index for A rows / N index for B cols
    const int khalf = (lane >> 4) << 1;   // 0 for lanes 0-15, 2 for lanes 16-31
    const int j0    = (blockIdx.y * 8 + wave) * 16;

    // Per-lane invariant bases; inner loop uses only compile-time offsets.
    const float* wpl   = W + n * (Kdim * Nout) + j0 + col + khalf * Nout;
    const int    abase = col * Kdim + khalf;

    v8f c = {};
    #pragma unroll 8
    for (int k0 = 0; k0 < Kdim; k0 += 4) {
        if ((k0 & 31) == 0 && (k0 + 32) < Kdim) {
            __builtin_prefetch(wpl + (k0 + 32) * Nout, 0, 1);
        }
        v2f a, b;
        // A 16x4 fp32 layout: lanes hold M, VGPR0={K+0,K+2}, VGPR1={K+1,K+3}
        a.x = As[abase + k0];
        a.y = As[abase + k0 + 1];
        // B 4x16 fp32 layout (mirrored): lanes hold N, halves/VGPRs hold K
        b.x = wpl[k0 * Nout];
        b.y = wpl[k0 * Nout + Nout];
        c = __builtin_amdgcn_wmma_f32_16x16x4_f32(
                /*neg_a=*/false, a, /*neg_b=*/false, b,
                /*c_mod=*/(short)0, c, /*reuse_a=*/false, /*reuse_b=*/false);
    }

    // C/D 16x16 fp32 layout: VGPR r holds M=r (lanes 0-15) / M=r+8 (lanes 16-31).
    // Batch rows 0..7 live entirely in the low half-wave.
    if (lane < 16) {
        const float bv    = bias[n * Nout + j0 + lane];
        float*      obase = out + n * OSN + j0 + lane;
        #pragma unroll
        for (int r = 0; r < 8; ++r) {
            float v = c[r] + bv;
            float* p = obase + r * OSB;
            if (ACC)  v += *p;
            if (RELU) v = fmaxf(v, 0.0f);
            *p = v;
        }
    }
}

// ---------------------------------------------------------------------------
// Fused attention: scores -> softmax -> context. One block per (b, n).
// Q/K/prev are ~2MB each -> L2-resident; VALU is plenty here.
// ---------------------------------------------------------------------------
__global__ __launch_bounds__(256) void attention_kernel(
    const float* __restrict__ Q, const float* __restrict__ Kmat,
    const float* __restrict__ prev, float* __restrict__ ctx)
{
    __shared__ float qs[KQ_];
    __shared__ float sc[N_];
    __shared__ float red[256];

    const int b = blockIdx.x / N_;
    const int n = blockIdx.x - b * N_;
    const int t = threadIdx.x;

    qs[t] = Q[((long)b * N_ + n) * KQ_ + t];
    __syncthreads();

    // thread t computes score against neuron m = t
    const float* krow = Kmat + ((long)b * N_ + t) * KQ_;
    float acc = 0.0f;
    #pragma unroll 8
    for (int k = 0; k < KQ_; ++k) acc += qs[k] * krow[k];
    const float score = acc * 0.0625f;     // 1/sqrt(KQ=256)

    // max-reduce
    red[t] = score; __syncthreads();
    for (int s = 128; s > 0; s >>= 1) {
        if (t < s) red[t] = fmaxf(red[t], red[t + s]);
        __syncthreads();
    }
    const float mx = red[0]; __syncthreads();

    const float e = __expf(score - mx);
    red[t] = e; __syncthreads();
    for (int s = 128; s > 0; s >>= 1) {
        if (t < s) red[t] += red[t + s];
        __syncthreads();
    }
    const float inv = 1.0f / red[0];
    sc[t] = e * inv;
    __syncthreads();

    // ctx[b,n,s=t] = sum_m w[m] * prev[b,m,t]   (coalesced across threads)
    const float* pb = prev + (long)b * N_ * S_ + t;
    float accs = 0.0f;
    #pragma unroll 8
    for (int m = 0; m < N_; ++m) accs += sc[m] * pb[m * S_];
    ctx[((long)b * N_ + n) * S_ + t] = accs;
}

// ---------------------------------------------------------------------------
// LSTM pointwise: gates [B,N,4H] + c -> c_new, h_new (in place)
// ---------------------------------------------------------------------------
__global__ __launch_bounds__(256) void lstm_pointwise(
    const float* __restrict__ gates, float* __restrict__ h, float* __restrict__ c)
{
    const long idx = (long)blockIdx.x * 256 + threadIdx.x;   // over B*N*H
    const long bn  = idx / H_;
    const int  hh  = (int)(idx - bn * H_);
    const float* g = gates + bn * (4 * H_);
    const float ig = g[hh];
    const float fg = g[H_ + hh];
    const float gg = g[2 * H_ + hh];
    const float og = g[3 * H_ + hh];
    const float si = 1.0f / (1.0f + __expf(-ig));
    const float sf = 1.0f / (1.0f + __expf(-fg));
    const float so = 1.0f / (1.0f + __expf(-og));
    const float cn = sf * c[idx] + si * tanhf(gg);
    c[idx] = cn;
    h[idx] = so * tanhf(cn);
}

// ---------------------------------------------------------------------------
// Output projection: out[b, t, o] = action[b,:] . w_proj[o,:] + b_proj[o]
// ---------------------------------------------------------------------------
__global__ __launch_bounds__(256) void proj_kernel(
    const float* __restrict__ prev, const float* __restrict__ wproj,
    const float* __restrict__ bproj, float* __restrict__ out, int tstep)
{
    __shared__ float red[256];
    const int o = blockIdx.x;
    const int b = blockIdx.y;
    const int t = threadIdx.x;
    const float* act = prev + ((long)b * N_ + (N_ - N_ACT_)) * S_;  // [8192] contig
    const float* w   = wproj + (long)o * (N_ACT_ * S_);
    float acc = 0.0f;
    #pragma unroll 4
    for (int k = t; k < N_ACT_ * S_; k += 256) acc += act[k] * w[k];
    red[t] = acc; __syncthreads();
    for (int s = 128; s > 0; s >>= 1) {
        if (t < s) red[t] += red[t + s];
        __syncthreads();
    }
    if (t == 0) out[((long)b * T_ + tstep) * OUTPUT_ + o] = red[0] + bproj[o];
}

__global__ void zero_kernel(float* __restrict__ p, long count)
{
    const long i = (long)blockIdx.x * blockDim.x + threadIdx.x;
    if (i < count) p[i] = 0.0f;
}

// ---------------------------------------------------------------------------
extern "C" void kernel_launch(void* const* d_in, const int* in_sizes, int n_in,
                              void* d_out, int out_size, void* d_ws, size_t ws_size,
                              hipStream_t stream)
{
    const float* x           = (const float*)d_in[0];
    const float* w_rec_input = (const float*)d_in[1];
    const float* b_rec_input = (const float*)d_in[2];
    const float* w_rec_other = (const float*)d_in[3];
    const float* b_rec_other = (const float*)d_in[4];
    const float* w_lstm_ih   = (const float*)d_in[5];
    const float* b_lstm_ih   = (const float*)d_in[6];
    const float* w_lstm_hh   = (const float*)d_in[7];
    const float* b_lstm_hh   = (const float*)d_in[8];
    const float* w_query     = (const float*)d_in[9];
    const float* b_query     = (const float*)d_in[10];
    const float* w_key       = (const float*)d_in[11];
    const float* b_key       = (const float*)d_in[12];
    const float* w_sender    = (const float*)d_in[13];
    const float* b_sender    = (const float*)d_in[14];
    const float* w_proj      = (const float*)d_in[15];
    const float* b_proj      = (const float*)d_in[16];
    float* outp = (float*)d_out;

    // Workspace layout (floats)
    float* ws      = (float*)d_ws;
    const long SZH = (long)B_ * N_ * H_;     // 524288
    float* h       = ws;
    float* c       = h    + SZH;
    float* prev    = c    + SZH;             // [B,N,S], S_==H_
    float* Qb      = prev + SZH;
    float* Kb      = Qb   + SZH;
    float* ctx     = Kb   + SZH;
    float* lstm_in = ctx  + SZH;
    float* gates   = lstm_in + SZH;          // [B,N,4H]

    // Deterministic zero state each call (h, c, prev are contiguous)
    {
        const long zn = 3 * SZH;
        zero_kernel<<<(unsigned)((zn + 255) / 256), 256, 0, stream>>>(h, zn);
    }

    constexpr int NH  = N_ * H_;          // 65536  (stride of h/Q/K/ctx/lstm_in per batch)
    constexpr int NG  = N_ * 4 * H_;      // 262144 (stride of gates per batch)

    for (int t = 0; t < T_; ++t) {
        // Q = h @ w_query + b_query ; K = h @ w_key + b_key
        neuron_gemm_wmma<H_, KQ_, NH, H_, NH, KQ_, false, false>
            <<<dim3(N_, KQ_ / 128), 256, 0, stream>>>(h, w_query, b_query, Qb);
        neuron_gemm_wmma<H_, KQ_, NH, H_, NH, KQ_, false, false>
            <<<dim3(N_, KQ_ / 128), 256, 0, stream>>>(h, w_key, b_key, Kb);

        // attention: scores -> softmax -> ctx (reads prev from the PREVIOUS step)
        attention_kernel<<<B_ * N_, 256, 0, stream>>>(Qb, Kb, prev, ctx);

        // rec_in = relu(x_t @ w_rec_input + b) -> lstm_in[:, 0:32, :]
        neuron_gemm_wmma<INPUT_, H_, T_ * INPUT_, 0, NH, H_, false, true>
            <<<dim3(N_IN_, H_ / 128), 256, 0, stream>>>(
                x + (long)t * INPUT_, w_rec_input, b_rec_input, lstm_in);

        // rec_oth = relu(ctx[:, 32:, :] @ w_rec_other + b) -> lstm_in[:, 32:, :]
        neuron_gemm_wmma<S_, H_, NH, S_, NH, H_, false, true>
            <<<dim3(N_TR_ + N_ACT_, H_ / 128), 256, 0, stream>>>(
                ctx + (long)N_IN_ * S_, w_rec_other, b_rec_other,
                lstm_in + (long)N_IN_ * H_);

        // gates = lstm_in @ w_ih + b_ih + h @ w_hh + b_hh   (two passes, acc)
        neuron_gemm_wmma<H_, 4 * H_, NH, H_, NG, 4 * H_, false, false>
            <<<dim3(N_, (4 * H_) / 128), 256, 0, stream>>>(
                lstm_in, w_lstm_ih, b_lstm_ih, gates);
        neuron_gemm_wmma<H_, 4 * H_, NH, H_, NG, 4 * H_, true, false>
            <<<dim3(N_, (4 * H_) / 128), 256, 0, stream>>>(
                h, w_lstm_hh, b_lstm_hh, gates);

        // c,h update in place (all readers of old h/c already ran this step)
        lstm_pointwise<<<(unsigned)(SZH / 256), 256, 0, stream>>>(gates, h, c);

        // prev = h_new @ w_sender + b_sender (prev already consumed by attention)
        neuron_gemm_wmma<H_, S_, NH, H_, NH, S_, false, false>
            <<<dim3(N_, S_ / 128), 256, 0, stream>>>(h, w_sender, b_sender, prev);

        // out[b, t, :] = action @ w_proj^T + b_proj
        proj_kernel<<<dim3(OUTPUT_, B_), 256, 0, stream>>>(prev, w_proj, b_proj, outp, t);
    }
}